// MultipolarRotation_7559142441141
// MI455X (gfx1250) — compile-verified
//
#include <hip/hip_runtime.h>

// d_out layout: [d : N*3 floats][q : N*9 floats]

struct F3 { float x, y, z; };

__device__ __forceinline__ F3 f3sub(F3 a, F3 b) { return {a.x - b.x, a.y - b.y, a.z - b.z}; }
__device__ __forceinline__ F3 f3add(F3 a, F3 b) { return {a.x + b.x, a.y + b.y, a.z + b.z}; }
__device__ __forceinline__ float f3dot(F3 a, F3 b) { return a.x * b.x + a.y * b.y + a.z * b.z; }
__device__ __forceinline__ F3 f3nrm(F3 v) {
    // reference: v / ||v||  -> v * rsqrt(dot(v,v)); v_rsq_f32 is a TRANS op (co-executes with VALU)
    float r = __builtin_amdgcn_rsqf(v.x * v.x + v.y * v.y + v.z * v.z);
    return {v.x * r, v.y * r, v.z * r};
}

// WGP-scope prefetch (scope field 0): pull the line into all cache levels,
// stopping on WGP hit. Addresses are always valid here (indices are % N),
// so the non-speculative WGP-scope behavior is safe. __builtin_prefetch
// only reaches SCOPE_SE, which stops at GL2 — useless since coords is
// already kept L2-resident by NT-tagging every other stream.
__device__ __forceinline__ void prefetch_wgp(const void* p) {
    asm volatile("global_prefetch_b8 %0, off" :: "v"(p));
}

__global__ void __launch_bounds__(256)
multipolar_rotation_kernel(const float* __restrict__ coords,
                           const int*   __restrict__ z_atoms,
                           const int*   __restrict__ x_atoms,
                           const int*   __restrict__ y_atoms,
                           const int*   __restrict__ axis_types,
                           const float* __restrict__ dipoles,
                           const float* __restrict__ quads,
                           float*       __restrict__ out_d,
                           float*       __restrict__ out_q,
                           int n)
{
    int i = blockIdx.x * blockDim.x + threadIdx.x;
    if (i >= n) return;

    const size_t i3 = 3 * (size_t)i;
    const size_t i9 = 9 * (size_t)i;

    // ---- index/axis streams are read-once: NT so they don't evict coords from L2 ----
    int zi = __builtin_nontemporal_load(z_atoms + i);
    int xi = __builtin_nontemporal_load(x_atoms + i);
    int yi = __builtin_nontemporal_load(y_atoms + i);
    int at = __builtin_nontemporal_load(axis_types + i);

    // ---- prefetch random gather lines into the WGP while the NT streams load ----
    const float* pz = coords + 3 * (size_t)zi;
    const float* px = coords + 3 * (size_t)xi;
    const float* py = coords + 3 * (size_t)yi;
    prefetch_wgp(pz);
    prefetch_wgp(px);
    prefetch_wgp(py);

    // ---- single-use streams: non-temporal so the 48MB coords table stays L2-resident ----
    float dpx = __builtin_nontemporal_load(dipoles + i3 + 0);
    float dpy = __builtin_nontemporal_load(dipoles + i3 + 1);
    float dpz = __builtin_nontemporal_load(dipoles + i3 + 2);
    float Q[9];
#pragma unroll
    for (int k = 0; k < 9; ++k) Q[k] = __builtin_nontemporal_load(quads + i9 + k);

    // ---- coords (RT-cached: the only reused array; gathers depend on its L2 residency) ----
    F3 c  = { coords[i3], coords[i3 + 1], coords[i3 + 2] };
    F3 zc = { pz[0], pz[1], pz[2] };
    F3 xc = { px[0], px[1], px[2] };
    F3 yc = { py[0], py[1], py[2] };

    // ---- rotation-frame construction (branchless; axis types mutually exclusive) ----
    F3 zv = f3nrm(f3sub(zc, c));
    F3 xg = f3nrm(f3sub(xc, c));

    bool fz_only = (at == 4) || (at == 5);
    F3 xz = { 1.0f - zv.x, zv.x, 0.0f };
    F3 xv = fz_only ? xz : xg;

    // bisector (axis_type == 1): z = norm(z + x)
    {
        F3 zb = f3nrm(f3add(zv, xv));
        zv = (at == 1) ? zb : zv;
    }

    F3 ynb = f3nrm(f3sub(yc, c));

    // z-then-bisector (axis_type == 2): x = norm(x + y)
    {
        F3 xb = f3nrm(f3add(xv, ynb));
        xv = (at == 2) ? xb : xv;
    }

    // 3-fold (axis_type == 3): z = norm(z + x + y)
    {
        F3 zt = f3nrm(f3add(f3add(zv, xv), ynb));
        zv = (at == 3) ? zt : zv;
    }

    // Gram-Schmidt: x = norm(x - z * dot(z, x))
    {
        float d = f3dot(zv, xv);
        F3 xo = { xv.x - zv.x * d, xv.y - zv.y * d, xv.z - zv.z * d };
        xv = f3nrm(xo);
    }

    // y = cross(z, x)
    F3 yv = { zv.y * xv.z - zv.z * xv.y,
              zv.z * xv.x - zv.x * xv.z,
              zv.x * xv.y - zv.y * xv.x };

    // no-axis (axis_type == 5): identity frame
    if (at == 5) { xv = {1.0f, 0.0f, 0.0f}; yv = {0.0f, 1.0f, 0.0f}; zv = {0.0f, 0.0f, 1.0f}; }

    // rot rows: r0 = xv, r1 = yv, r2 = zv
    // d_j = dip_x * xv[j] + dip_y * yv[j] + dip_z * zv[j]
    float d0 = dpx * xv.x + dpy * yv.x + dpz * zv.x;
    float d1 = dpx * xv.y + dpy * yv.y + dpz * zv.y;
    float d2 = dpx * xv.z + dpy * yv.z + dpz * zv.z;

    // t[k][j] = sum_l Q[k][l] * r_l[j]
    F3 t0 = { Q[0] * xv.x + Q[1] * yv.x + Q[2] * zv.x,
              Q[0] * xv.y + Q[1] * yv.y + Q[2] * zv.y,
              Q[0] * xv.z + Q[1] * yv.z + Q[2] * zv.z };
    F3 t1 = { Q[3] * xv.x + Q[4] * yv.x + Q[5] * zv.x,
              Q[3] * xv.y + Q[4] * yv.y + Q[5] * zv.y,
              Q[3] * xv.z + Q[4] * yv.z + Q[5] * zv.z };
    F3 t2 = { Q[6] * xv.x + Q[7] * yv.x + Q[8] * zv.x,
              Q[6] * xv.y + Q[7] * yv.y + Q[8] * zv.y,
              Q[6] * xv.z + Q[7] * yv.z + Q[8] * zv.z };

    // q[i][j] = r0[i]*t0[j] + r1[i]*t1[j] + r2[i]*t2[j]   (i,j over components)
    float q00 = xv.x * t0.x + yv.x * t1.x + zv.x * t2.x;
    float q01 = xv.x * t0.y + yv.x * t1.y + zv.x * t2.y;
    float q02 = xv.x * t0.z + yv.x * t1.z + zv.x * t2.z;
    float q10 = xv.y * t0.x + yv.y * t1.x + zv.y * t2.x;
    float q11 = xv.y * t0.y + yv.y * t1.y + zv.y * t2.y;
    float q12 = xv.y * t0.z + yv.y * t1.z + zv.y * t2.z;
    float q20 = xv.z * t0.x + yv.z * t1.x + zv.z * t2.x;
    float q21 = xv.z * t0.y + yv.z * t1.y + zv.z * t2.y;
    float q22 = xv.z * t0.z + yv.z * t1.z + zv.z * t2.z;

    // ---- non-temporal stores: 192MB write-once stream, keep it out of L2 ----
    __builtin_nontemporal_store(d0, out_d + i3 + 0);
    __builtin_nontemporal_store(d1, out_d + i3 + 1);
    __builtin_nontemporal_store(d2, out_d + i3 + 2);

    __builtin_nontemporal_store(q00, out_q + i9 + 0);
    __builtin_nontemporal_store(q01, out_q + i9 + 1);
    __builtin_nontemporal_store(q02, out_q + i9 + 2);
    __builtin_nontemporal_store(q10, out_q + i9 + 3);
    __builtin_nontemporal_store(q11, out_q + i9 + 4);
    __builtin_nontemporal_store(q12, out_q + i9 + 5);
    __builtin_nontemporal_store(q20, out_q + i9 + 6);
    __builtin_nontemporal_store(q21, out_q + i9 + 7);
    __builtin_nontemporal_store(q22, out_q + i9 + 8);
}

extern "C" void kernel_launch(void* const* d_in, const int* in_sizes, int n_in,
                              void* d_out, int out_size, void* d_ws, size_t ws_size,
                              hipStream_t stream) {
    const float* coords     = (const float*)d_in[0];
    const int*   z_atoms    = (const int*)d_in[1];
    const int*   x_atoms    = (const int*)d_in[2];
    const int*   y_atoms    = (const int*)d_in[3];
    const int*   axis_types = (const int*)d_in[4];
    const float* dipoles    = (const float*)d_in[5];
    const float* quads      = (const float*)d_in[6];

    const int n = in_sizes[1];  // z_atoms count == N

    float* out_d = (float*)d_out;           // N*3
    float* out_q = out_d + 3 * (size_t)n;   // N*9

    const int threads = 256;  // 8 wave32 waves per block
    const int blocks  = (n + threads - 1) / threads;
    multipolar_rotation_kernel<<<blocks, threads, 0, stream>>>(
        coords, z_atoms, x_atoms, y_atoms, axis_types, dipoles, quads,
        out_d, out_q, n);
}